// TorchFEMMesh_5007931867268
// MI455X (gfx1250) — compile-verified
//
#include <hip/hip_runtime.h>
#include <math.h>

// FEM conductivity solve for MI455X (gfx1250, wave32).
// Dominant cost: dense SPD solve (9024x9024) -> blocked Cholesky, outer block
// 32, with panel/trailing GEMMs on V_WMMA_F32_16X16X4_F32. Trailing update
// uses 2x2 register tiling (32x32 macro-tile per wave, 32 WMMA, K=32).

typedef float v2f __attribute__((ext_vector_type(2)));
typedef float v8f __attribute__((ext_vector_type(8)));

#define LOW_COND 0.001f
#define KB 32  // outer Cholesky block

// ---------------- WMMA tile helpers (wave32, 16x16 f32) ----------------
// A 16x4 f32 (ISA 7.12.2): lane L holds A[M=L%16][K=2*(L/16)+v] in VGPR v.
// B 4x16: lane L holds B[K=2*(L/16)+v][N=L%16] in VGPR v.
// C/D 16x16: lane L = 16*(m>>3) + n holds D[m][n] in VGPR (m&7).

__device__ inline v8f wmma4(v2f a, v2f b, v8f c) {
  return __builtin_amdgcn_wmma_f32_16x16x4_f32(false, a, false, b, (short)0, c,
                                               false, false);
}

// Row fragment: element pair (row m, cols k4..k4+1) of a row-major tile.
// Serves both the A operand and a transposed-B operand (B = tile^T).
__device__ inline v2f ldfrag(const float* __restrict__ p, long ld, int m, int k4) {
  v2f r;
  r.x = p[(long)m * ld + k4];
  r.y = p[(long)m * ld + k4 + 1];
  return r;
}

__device__ inline v8f load_C16(const float* __restrict__ C, long ldc) {
  const int lane = threadIdx.x & 31;
  const int n = lane & 15;
  const int half = lane >> 4;
  v8f c;
#pragma unroll
  for (int r = 0; r < 8; ++r) c[r] = C[(long)(r + 8 * half) * ldc + n];
  return c;
}

__device__ inline void store_C16(float* __restrict__ C, long ldc, v8f c) {
  const int lane = threadIdx.x & 31;
  const int n = lane & 15;
  const int half = lane >> 4;
#pragma unroll
  for (int r = 0; r < 8; ++r) C[(long)(r + 8 * half) * ldc + n] = c[r];
}

// ---------------- setup / assembly kernels ----------------

__global__ void k_zeroK4(float4* __restrict__ K, long tot4) {
  long i = (long)blockIdx.x * blockDim.x + threadIdx.x;
  const long stride = (long)gridDim.x * blockDim.x;
  const float4 z = make_float4(0.f, 0.f, 0.f, 0.f);
  for (; i < tot4; i += stride) K[i] = z;
}

__global__ void k_paddiag(float* __restrict__ K, long npad, int nf, int npi) {
  int i = blockIdx.x * blockDim.x + threadIdx.x;
  if (i < npi) {
    long r = nf + i;
    K[r * npad + r] = 1.0f;  // identity on padded rows -> x_pad = 0
  }
}

__global__ void k_init_misc(int* __restrict__ map_free, float* __restrict__ b,
                            float* __restrict__ out, int nn, long npad) {
  long i = (long)blockIdx.x * blockDim.x + threadIdx.x;
  const long stride = (long)gridDim.x * blockDim.x;
  long tot = (nn > npad) ? nn : npad;
  for (; i < tot; i += stride) {
    if (i < nn) map_free[i] = -1;
    if (i < npad) b[i] = 0.0f;
    if (i == 0) out[0] = 0.0f;
  }
}

__global__ void k_scatter_maps(const int* __restrict__ fc, const int* __restrict__ bc,
                               int* __restrict__ map_free, int* __restrict__ map_bc,
                               int nf, int nbc) {
  int i = blockIdx.x * blockDim.x + threadIdx.x;
  if (i < nf) map_free[fc[i]] = i;
  if (i < nbc) map_bc[bc[i]] = i;
}

__global__ void k_assemble(const float* __restrict__ base, const int* __restrict__ eidx,
                           const int* __restrict__ rows, const int* __restrict__ cols,
                           const float* __restrict__ mask, const float* __restrict__ u_c,
                           const int* __restrict__ map_free, const int* __restrict__ map_bc,
                           float* __restrict__ K, float* __restrict__ b,
                           long npad, int n9e) {
  int i = blockIdx.x * blockDim.x + threadIdx.x;
  if (i >= n9e) return;
  int e = eidx[i];
  float sigma = LOW_COND + (1.0f - LOW_COND) * mask[e >> 1];
  float val = base[i] * sigma;
  int r = rows[i], c = cols[i];
  int fr = map_free[r];
  if (fr < 0) return;  // constrained row: dropped
  int fcc = map_free[c];
  if (fcc >= 0) {
    atomicAdd(&K[(long)fr * npad + fcc], val);  // K_ff
  } else {
    atomicAdd(&b[fr], -val * u_c[map_bc[c]]);   // b_f = -K_fc @ u_c
  }
}

// ---------------- blocked Cholesky (outer block KB=32) ----------------

// 32x32 diagonal Cholesky + explicit lower-triangular inverse (serial, tiny).
__global__ void k_potrf(float* __restrict__ K, float* __restrict__ invL,
                        long npad, int kb) {
  __shared__ float L[KB][KB];
  __shared__ float iv[KB][KB];
  if (threadIdx.x == 0) {
    float* D = K + (long)kb * KB * npad + (long)kb * KB;
    for (int r = 0; r < KB; ++r)
      for (int c = 0; c <= r; ++c) L[r][c] = D[(long)r * npad + c];
    for (int j = 0; j < KB; ++j) {
      float s = L[j][j];
      for (int c = 0; c < j; ++c) s -= L[j][c] * L[j][c];
      float d = sqrtf(fmaxf(s, 1e-30f));
      L[j][j] = d;
      float inv = 1.0f / d;
      for (int r = j + 1; r < KB; ++r) {
        float s2 = L[r][j];
        for (int c = 0; c < j; ++c) s2 -= L[r][c] * L[j][c];
        L[r][j] = s2 * inv;
      }
    }
    // write back L (zero strict upper), then invert by forward substitution
    for (int r = 0; r < KB; ++r)
      for (int c = 0; c < KB; ++c)
        D[(long)r * npad + c] = (c <= r) ? L[r][c] : 0.0f;
    for (int j = 0; j < KB; ++j)
      for (int i = 0; i < KB; ++i) {
        if (i < j) { iv[i][j] = 0.0f; continue; }
        float s = (i == j) ? 1.0f : 0.0f;
        for (int c = j; c < i; ++c) s -= L[i][c] * iv[c][j];
        iv[i][j] = s / L[i][i];
      }
    for (int r = 0; r < KB; ++r)
      for (int c = 0; c < KB; ++c) invL[r * KB + c] = iv[r][c];
  }
}

// Panel: L_rt = A_rt(16x32) * invL^T(32x32). One wave per 16-row tile,
// shared A fragment across the two output column tiles: 16 WMMA per wave.
__global__ void k_panel(float* __restrict__ K, const float* __restrict__ invL,
                        long npad, int kb) {
  long row0 = ((long)(kb + 1) * 2 + blockIdx.x) * 16;
  float* A = K + row0 * npad + (long)kb * KB;
  const int lane = threadIdx.x & 31;
  const int m = lane & 15;
  const int half = lane >> 4;
  v8f c0 = {0.f, 0.f, 0.f, 0.f, 0.f, 0.f, 0.f, 0.f};
  v8f c1 = c0;
#pragma unroll
  for (int t = 0; t < 8; ++t) {
    const int k4 = 4 * t + 2 * half;
    v2f a  = ldfrag(A, npad, m, k4);
    v2f b0 = ldfrag(invL, KB, m, k4);            // out cols 0..15
    v2f b1 = ldfrag(invL + 16 * KB, KB, m, k4);  // out cols 16..31
    c0 = wmma4(a, b0, c0);
    c1 = wmma4(a, b1, c1);
  }
  store_C16(A, npad, c0);
  store_C16(A + 16, npad, c1);
}

// Trailing SYRK: C_IJ(32x32) -= L_I(32x32) * L_J(32x32)^T for kb < J <= I.
// 256 threads = 8 waves/block; each wave owns one 32x32 macro-tile:
// 2x2 register tiling, K=32 -> 32 WMMA per wave, fragments reused twice.
// Diagonal macro-tiles also update their (symmetric) upper 16-block, which
// keeps the code branch-free (EXEC all-ones at every WMMA) and is harmless.
__global__ void k_trail(float* __restrict__ K, long npad, int kb, int nrem) {
  int wave = threadIdx.x >> 5;
  int I = kb + 1 + blockIdx.y;
  int J = kb + 1 + blockIdx.x * 8 + wave;
  if (J > I || J > kb + nrem) return;  // triangular-only (wave-uniform exit)
  const int lane = threadIdx.x & 31;
  const int m = lane & 15;
  const int half = lane >> 4;
  const float* Li0 = K + (long)(2 * I) * 16 * npad + (long)kb * KB;
  const float* Li1 = Li0 + (long)16 * npad;
  const float* Lj0 = K + (long)(2 * J) * 16 * npad + (long)kb * KB;
  const float* Lj1 = Lj0 + (long)16 * npad;
  float* C00 = K + (long)(2 * I) * 16 * npad + (long)(2 * J) * 16;
  float* C01 = C00 + 16;
  float* C10 = C00 + (long)16 * npad;
  float* C11 = C10 + 16;
  v8f c00 = load_C16(C00, npad);
  v8f c01 = load_C16(C01, npad);
  v8f c10 = load_C16(C10, npad);
  v8f c11 = load_C16(C11, npad);
#pragma unroll
  for (int t = 0; t < 8; ++t) {
    const int k4 = 4 * t + 2 * half;
    v2f ai0 = ldfrag(Li0, npad, m, k4);
    v2f ai1 = ldfrag(Li1, npad, m, k4);
    // f32 WMMA NEG modifier only covers C -> negate A in VALU for "-="
    ai0.x = -ai0.x; ai0.y = -ai0.y;
    ai1.x = -ai1.x; ai1.y = -ai1.y;
    v2f bj0 = ldfrag(Lj0, npad, m, k4);
    v2f bj1 = ldfrag(Lj1, npad, m, k4);
    c00 = wmma4(ai0, bj0, c00);
    c01 = wmma4(ai0, bj1, c01);
    c10 = wmma4(ai1, bj0, c10);
    c11 = wmma4(ai1, bj1, c11);
  }
  store_C16(C00, npad, c00);
  store_C16(C01, npad, c01);
  store_C16(C10, npad, c10);
  store_C16(C11, npad, c11);
}

// ---------------- triangular solves (O(n^2), single workgroup) ----------------

__global__ void k_fwd(const float* __restrict__ K, float* __restrict__ b,
                      long npad, int nt16) {
  __shared__ float ys[16];
  for (int kb = 0; kb < nt16; ++kb) {
    const float* Lk = K + (long)kb * 16 * npad + (long)kb * 16;
    if (threadIdx.x == 0) {
      for (int r = 0; r < 16; ++r) {
        float s = b[kb * 16 + r];
        for (int c = 0; c < r; ++c) s -= Lk[(long)r * npad + c] * ys[c];
        ys[r] = s / Lk[(long)r * npad + r];
      }
    }
    __syncthreads();
    if (threadIdx.x < 16) b[kb * 16 + threadIdx.x] = ys[threadIdx.x];
    for (long i = (long)(kb + 1) * 16 + threadIdx.x; i < npad; i += blockDim.x) {
      const float* Lrow = K + i * npad + (long)kb * 16;
      float s = 0.0f;
#pragma unroll
      for (int c = 0; c < 16; ++c) s += Lrow[c] * ys[c];
      b[i] -= s;
    }
    __syncthreads();
  }
}

__global__ void k_bwd(const float* __restrict__ K, float* __restrict__ b,
                      long npad, int nt16) {
  __shared__ float xs[16];
  for (int kb = nt16 - 1; kb >= 0; --kb) {
    const float* Lk = K + (long)kb * 16 * npad + (long)kb * 16;
    if (threadIdx.x == 0) {
      for (int r = 15; r >= 0; --r) {
        float s = b[kb * 16 + r];
        for (int c = r + 1; c < 16; ++c) s -= Lk[(long)c * npad + r] * xs[c];
        xs[r] = s / Lk[(long)r * npad + r];
      }
    }
    __syncthreads();
    if (threadIdx.x < 16) b[kb * 16 + threadIdx.x] = xs[threadIdx.x];
    for (long i = threadIdx.x; i < (long)kb * 16; i += blockDim.x) {
      float s = 0.0f;
#pragma unroll
      for (int c = 0; c < 16; ++c) s += K[((long)kb * 16 + c) * npad + i] * xs[c];
      b[i] -= s;
    }
    __syncthreads();
  }
}

// ---------------- solution scatter + energy functional ----------------

__global__ void k_scatter_u(const int* __restrict__ map_free, const int* __restrict__ map_bc,
                            const float* __restrict__ uf, const float* __restrict__ u_c,
                            float* __restrict__ u, int nn) {
  int i = blockIdx.x * blockDim.x + threadIdx.x;
  if (i >= nn) return;
  int f = map_free[i];
  u[i] = (f >= 0) ? uf[f] : u_c[map_bc[i]];
}

__global__ void k_energy(const int* __restrict__ elems, const float* __restrict__ gT,
                         const float* __restrict__ area, const float* __restrict__ mask,
                         const float* __restrict__ u, float* __restrict__ out, int E) {
  __shared__ float red[256];
  int e = blockIdx.x * blockDim.x + threadIdx.x;
  float v = 0.0f;
  if (e < E) {
    float u0 = u[elems[3 * e]];
    float u1 = u[elems[3 * e + 1]];
    float u2 = u[elems[3 * e + 2]];
    const float* g = gT + 6 * e;
    float g0 = g[0] * u0 + g[1] * u1 + g[2] * u2;
    float g1 = g[3] * u0 + g[4] * u1 + g[5] * u2;
    float sigma = LOW_COND + (1.0f - LOW_COND) * mask[e >> 1];
    v = sigma * (g0 * g0 + g1 * g1) * area[e];
  }
  red[threadIdx.x] = v;
  __syncthreads();
  for (int s = 128; s > 0; s >>= 1) {
    if (threadIdx.x < s) red[threadIdx.x] += red[threadIdx.x + s];
    __syncthreads();
  }
  if (threadIdx.x == 0) atomicAdd(out, red[0]);
}

// ---------------- host orchestration ----------------

extern "C" void kernel_launch(void* const* d_in, const int* in_sizes, int n_in,
                              void* d_out, int out_size, void* d_ws, size_t ws_size,
                              hipStream_t stream) {
  (void)n_in; (void)out_size; (void)ws_size;
  // setup_inputs() dict order:
  const int*   elems = (const int*)d_in[0];
  const int*   rows  = (const int*)d_in[1];
  const int*   cols  = (const int*)d_in[2];
  const float* base  = (const float*)d_in[3];
  const int*   eidx  = (const int*)d_in[4];
  const float* gT    = (const float*)d_in[5];
  const float* area  = (const float*)d_in[6];
  const int*   bc    = (const int*)d_in[7];
  const int*   fc    = (const int*)d_in[8];
  const float* u_c   = (const float*)d_in[9];
  const float* mask  = (const float*)d_in[10];

  const int E    = in_sizes[6];   // elem_area: [E]
  const int n9e  = in_sizes[3];   // base_data: [9E]
  const int Nbc  = in_sizes[7];
  const int Nf   = in_sizes[8];
  const int nn   = Nf + Nbc;
  const int NB2  = (Nf + KB - 1) / KB;  // 32-wide outer blocks
  const long npad = (long)NB2 * KB;
  const int nt16 = (int)(npad / 16);

  char* ws = (char*)d_ws;
  float* K        = (float*)ws;  ws += sizeof(float) * npad * npad;
  float* b        = (float*)ws;  ws += sizeof(float) * npad;
  float* u        = (float*)ws;  ws += sizeof(float) * nn;
  float* invL     = (float*)ws;  ws += sizeof(float) * KB * KB;
  int*   map_free = (int*)ws;    ws += sizeof(int) * nn;
  int*   map_bc   = (int*)ws;

  // --- init + assembly ---
  k_zeroK4<<<4096, 256, 0, stream>>>((float4*)K, npad * npad / 4);
  if (npad > Nf) {
    int npi = (int)(npad - Nf);
    k_paddiag<<<(npi + 255) / 256, 256, 0, stream>>>(K, npad, Nf, npi);
  }
  k_init_misc<<<256, 256, 0, stream>>>(map_free, b, (float*)d_out, nn, npad);
  int mx = (Nf > Nbc) ? Nf : Nbc;
  k_scatter_maps<<<(mx + 255) / 256, 256, 0, stream>>>(fc, bc, map_free, map_bc, Nf, Nbc);
  k_assemble<<<(n9e + 255) / 256, 256, 0, stream>>>(base, eidx, rows, cols, mask, u_c,
                                                    map_free, map_bc, K, b, npad, n9e);

  // --- blocked Cholesky (WMMA panel + trailing update) ---
  for (int kb = 0; kb < NB2; ++kb) {
    k_potrf<<<1, 32, 0, stream>>>(K, invL, npad, kb);
    int nrem = NB2 - 1 - kb;
    if (nrem > 0) {
      k_panel<<<2 * nrem, 32, 0, stream>>>(K, invL, npad, kb);
      dim3 g((nrem + 7) / 8, nrem);
      k_trail<<<g, 256, 0, stream>>>(K, npad, kb, nrem);
    }
  }

  // --- solves + energy ---
  k_fwd<<<1, 256, 0, stream>>>(K, b, npad, nt16);
  k_bwd<<<1, 256, 0, stream>>>(K, b, npad, nt16);
  k_scatter_u<<<(nn + 255) / 256, 256, 0, stream>>>(map_free, map_bc, b, u_c, u, nn);
  k_energy<<<(E + 255) / 256, 256, 0, stream>>>(elems, gT, area, mask, u, (float*)d_out, E);
}